// PoseCoTrain_18854906430077
// MI455X (gfx1250) — compile-verified
//
#include <hip/hip_runtime.h>

typedef __attribute__((ext_vector_type(2))) float v2f;
typedef __attribute__((ext_vector_type(8))) float v8f;

// Per-wave affine transform via V_WMMA_F32_16X16X4_F32.
// A (16x4): row0=(t00,t01,0,0), row1=(t10,t11,0,0), row8=(0,0,t00,t01), row9=(0,0,t10,t11).
//   A VGPR layout: v0 = K0 (lanes 0-15) / K2 (lanes 16-31); v1 = K1 / K3
//   -> nonzero only in lanes 0,1,24,25.
// B (4x16): v0 = K0 row (lanes 0-15) / K2 row (lanes 16-31); v1 = K1 / K3
//   -> setting v0 = x_t, v1 = y_t in EVERY lane supplies each lane's own coords.
// C: row0/row8 = t02, row1/row9 = t12 (C v0/v1 broadcast) adds translation.
// D: v0 = gx(lane), v1 = gy(lane) for all 32 lanes.
__device__ __forceinline__ void affine_wmma(float t00, float t01, float t02,
                                            float t10, float t11, float t12,
                                            float xt, float yt,
                                            float& gx, float& gy) {
    const int lane = threadIdx.x & 31;
    const bool r0 = (lane == 0) | (lane == 24);
    const bool r1 = (lane == 1) | (lane == 25);
    v2f A;
    A.x = r0 ? t00 : (r1 ? t10 : 0.0f);
    A.y = r0 ? t01 : (r1 ? t11 : 0.0f);
    v2f B;
    B.x = xt;
    B.y = yt;
    v8f C = {};
    C[0] = t02;
    C[1] = t12;
    v8f D = __builtin_amdgcn_wmma_f32_16x16x4_f32(
        /*neg_a=*/false, A, /*neg_b=*/false, B,
        /*c_mod=*/(short)0, C, /*reuse_a=*/false, /*reuse_b=*/false);
    gx = D[0];
    gy = D[1];
}

// Exact replication of the reference bilinear weight/index math.
// Note: reference scales by (dim-2), clips x1 from UNclipped x0, then clips x0.
__device__ __forceinline__ void bilinear_setup(float gx, float gy, int W, int H,
                                               int& x0c, int& x1, int& y0c, int& y1,
                                               float& wa, float& wb, float& wc, float& wd) {
    const float xf = 0.5f * (gx + 1.0f) * (float)(W - 2);
    const float yf = 0.5f * (gy + 1.0f) * (float)(H - 2);
    const int x0 = (int)floorf(xf);
    const int y0 = (int)floorf(yf);
    x1  = min(max(x0 + 1, 0), W - 1);
    y1  = min(max(y0 + 1, 0), H - 1);
    x0c = min(max(x0, 0), W - 1);
    y0c = min(max(y0, 0), H - 1);
    const float x0f = (float)x0c, x1f = (float)x1;
    const float y0f = (float)y0c, y1f = (float)y1;
    wa = (x1f - xf) * (y1f - yf);
    wb = (x1f - xf) * (yf - y0f);
    wc = (xf - x0f) * (y1f - yf);
    wd = (xf - x0f) * (yf - y0f);
}

// Image: 128 x 256 x 192 x 3. One thread per output pixel (3 channels).
// Stores: 12 contiguous bytes/thread -> 384B/wave, fully coalesced.
__global__ __launch_bounds__(256) void stn_img_kernel(const float* __restrict__ img,
                                                      const float* __restrict__ thetas,
                                                      float* __restrict__ out) {
    constexpr int H = 256, W = 192, C = 3;
    const int b = blockIdx.y;
    const int p = blockIdx.x * 256 + threadIdx.x;   // 0..49151, exact
    const int iy = p / W;
    const int ix = p - iy * W;

    const float* t = thetas + b * 6;
    const float t00 = t[0], t01 = t[1], t02 = t[2];
    const float t10 = t[3], t11 = t[4], t12 = t[5];

    const float xt = 2.0f * (float)ix / (float)(W - 1) - 1.0f;
    const float yt = 2.0f * (float)iy / (float)(H - 1) - 1.0f;

    float gx, gy;
    affine_wmma(t00, t01, t02, t10, t11, t12, xt, yt, gx, gy);

    int x0c, x1, y0c, y1;
    float wa, wb, wc, wd;
    bilinear_setup(gx, gy, W, H, x0c, x1, y0c, y1, wa, wb, wc, wd);

    const float* base = img + (size_t)b * H * W * C;
    const float* pa = base + (size_t)(y0c * W + x0c) * C;
    const float* pb = base + (size_t)(y1  * W + x0c) * C;
    const float* pc = base + (size_t)(y0c * W + x1 ) * C;
    const float* pd = base + (size_t)(y1  * W + x1 ) * C;

    float* o = out + ((size_t)b * H * W + p) * C;
#pragma unroll
    for (int c = 0; c < C; ++c) {
        o[c] = wa * pa[c] + wb * pb[c] + wc * pc[c] + wd * pd[c];
    }
}

// Heatmap: 128 x 64 x 48 x 17. One thread per output ELEMENT (pixel,channel):
// consecutive threads write consecutive floats (coalesced); 17 lanes sharing a
// pixel also read 17 contiguous floats per corner (coalesced gathers via L2).
__global__ __launch_bounds__(256) void stn_hm_kernel(const float* __restrict__ hm,
                                                     const float* __restrict__ thetas,
                                                     float* __restrict__ out) {
    constexpr int H = 64, W = 48, C = 17;
    const int b = blockIdx.y;
    const int e = blockIdx.x * 256 + threadIdx.x;   // 0..52223, exact
    const int p = e / C;
    const int c = e - p * C;
    const int iy = p / W;
    const int ix = p - iy * W;

    const float* t = thetas + b * 6;
    const float t00 = t[0], t01 = t[1], t02 = t[2];
    const float t10 = t[3], t11 = t[4], t12 = t[5];

    const float xt = 2.0f * (float)ix / (float)(W - 1) - 1.0f;
    const float yt = 2.0f * (float)iy / (float)(H - 1) - 1.0f;

    float gx, gy;
    affine_wmma(t00, t01, t02, t10, t11, t12, xt, yt, gx, gy);

    int x0c, x1, y0c, y1;
    float wa, wb, wc, wd;
    bilinear_setup(gx, gy, W, H, x0c, x1, y0c, y1, wa, wb, wc, wd);

    const float* base = hm + (size_t)b * H * W * C;
    const float va = base[(size_t)(y0c * W + x0c) * C + c];
    const float vb = base[(size_t)(y1  * W + x0c) * C + c];
    const float vc = base[(size_t)(y0c * W + x1 ) * C + c];
    const float vd = base[(size_t)(y1  * W + x1 ) * C + c];

    out[(size_t)b * H * W * C + e] = wa * va + wb * vb + wc * vc + wd * vd;
}

extern "C" void kernel_launch(void* const* d_in, const int* in_sizes, int n_in,
                              void* d_out, int out_size, void* d_ws, size_t ws_size,
                              hipStream_t stream) {
    (void)in_sizes; (void)n_in; (void)out_size; (void)d_ws; (void)ws_size;
    const float* img    = (const float*)d_in[0];  // (128,256,192,3) f32
    const float* hm     = (const float*)d_in[1];  // (128,64,48,17)  f32
    const float* thetas = (const float*)d_in[2];  // (128,2,3)       f32

    float* out_img = (float*)d_out;
    float* out_hm  = out_img + (size_t)128 * 256 * 192 * 3;  // tuple order: (img_out, hm_out)

    // img: 49152 px/batch / 256 = 192 blocks; hm: 52224 elem/batch / 256 = 204 blocks.
    dim3 gi(192, 128), gh(204, 128);
    stn_img_kernel<<<gi, 256, 0, stream>>>(img, thetas, out_img);
    stn_hm_kernel <<<gh, 256, 0, stream>>>(hm,  thetas, out_hm);
}